// GCN_MLP_31172872634622
// MI455X (gfx1250) — compile-verified
//
#include <hip/hip_runtime.h>
#include <hip/hip_bf16.h>

#define NN 100000
#define NE 1600000
#define NL 1000000
#define IND 128
#define HID 256
#define OUTD 128

typedef float v2f __attribute__((ext_vector_type(2)));
typedef float v8f __attribute__((ext_vector_type(8)));

// ---------------------------------------------------------------- utilities
__global__ void zero_f32(float* p, long n) {
    long i = (long)blockIdx.x * blockDim.x + threadIdx.x;
    if (i < n) p[i] = 0.0f;
}

__global__ void count_deg(const int* __restrict__ dst, float* deg, int E) {
    int i = blockIdx.x * blockDim.x + threadIdx.x;
    if (i < E) atomicAdd(&deg[dst[i]], 1.0f);
}

__global__ void finalize_deg(const float* __restrict__ deg,
                             float* __restrict__ dinv,
                             float* __restrict__ invdeg, int n) {
    int i = blockIdx.x * blockDim.x + threadIdx.x;
    if (i < n) {
        float d = deg[i] + 1.0f;          // +1 self-loop
        dinv[i]   = rsqrtf(d);
        invdeg[i] = 1.0f / d;
    }
}

// ---------------------------------------------------------------- dense GEMM: C = A @ B (f32 WMMA)
// one wave -> 16 rows x 64 cols (4 n-tiles); A fragment reused across the 4 WMMAs per k-step
__global__ void gemm_wmma(const float* __restrict__ A, const float* __restrict__ B,
                          float* __restrict__ C, int M, int N, int K) {
    int wave  = (blockIdx.x * blockDim.x + threadIdx.x) >> 5;   // uniform per wave
    int tn    = N >> 6;                                         // 64-col groups
    int total = (M >> 4) * tn;
    if (wave >= total) return;                                  // wave-uniform exit
    int m0 = (wave / tn) << 4;
    int n0 = (wave % tn) << 6;
    int lane = threadIdx.x & 31;
    int half = lane >> 4;
    int r    = lane & 15;

    v8f acc[4] = {v8f{}, v8f{}, v8f{}, v8f{}};
    const float* Arow = A + (long)(m0 + r) * K;
    for (int k = 0; k < K; k += 4) {
        int ka = k + 2 * half;            // A 16x4 layout: v0={K0|K2}, v1={K1|K3}
        v2f a;
        a[0] = Arow[ka];
        a[1] = Arow[ka + 1];
        const float* B0 = B + (long)ka * N + n0 + r;
        const float* B1 = B + (long)(ka + 1) * N + n0 + r;
#pragma unroll
        for (int nt = 0; nt < 4; ++nt) {
            v2f b;
            b[0] = B0[nt * 16];
            b[1] = B1[nt * 16];
            acc[nt] = __builtin_amdgcn_wmma_f32_16x16x4_f32(false, a, false, b,
                                                            (short)0, acc[nt], false, false);
        }
    }
    float* Crow = C + (long)(m0 + 8 * half) * N + n0 + r;       // D: vgpr i -> row m0+i+8*half
#pragma unroll
    for (int nt = 0; nt < 4; ++nt)
#pragma unroll
        for (int i = 0; i < 8; ++i) Crow[(long)i * N + nt * 16] = acc[nt][i];
}

// ---------------------------------------------------------------- edge aggregation (scatter-add)
__global__ void edge_scatter(const float* __restrict__ H,
                             const int* __restrict__ src, const int* __restrict__ dst,
                             const float* __restrict__ dinv,
                             float* __restrict__ AGG, int E, int D) {
    long t   = (long)blockIdx.x * blockDim.x + threadIdx.x;
    int  grp = D >> 2;
    if (t >= (long)E * grp) return;
    int e = (int)(t / grp);
    int j = (int)(t % grp) << 2;
    int s = src[e], d = dst[e];
    float nrm = dinv[s] * dinv[d];
    const float4 v = *(const float4*)(H + (long)s * D + j);
    float* o = AGG + (long)d * D + j;
    atomicAdd(o + 0, v.x * nrm);
    atomicAdd(o + 1, v.y * nrm);
    atomicAdd(o + 2, v.z * nrm);
    atomicAdd(o + 3, v.w * nrm);
}

// ---------------------------------------------------------------- self-loop + bias (+ relu), in place
__global__ void post_conv(float* __restrict__ AGG, const float* __restrict__ H,
                          const float* __restrict__ invdeg, const float* __restrict__ bias,
                          long n, int D, int do_relu) {
    long t = (long)blockIdx.x * blockDim.x + threadIdx.x;
    if (t >= n * D) return;
    long i = t / D;
    int  d = (int)(t % D);
    float v = AGG[t] + H[t] * invdeg[i] + bias[d];
    if (do_relu) v = fmaxf(v, 0.0f);
    AGG[t] = v;
}

// ---------------------------------------------------------------- fused decoder (LDS-staged W3)
// Block = 8 waves. The block stages all of W3 (256KB) into LDS (CDNA5: 320KB/WGP) as
// interleaved K-pairs so each B fragment is one ds_load_b64. One wave owns 16 label-edge
// rows and holds all 16 n-tile accumulators (128 VGPRs), so each gathered Z2 row is read
// from L2 exactly once. relu+b3+W4-dot folded in registers; H3 never touches memory.
__global__ void decode_wmma(const float* __restrict__ Z,
                            const int* __restrict__ li0, const int* __restrict__ li1,
                            const float* __restrict__ W3, const float* __restrict__ b3,
                            const float* __restrict__ W4, const float* __restrict__ b4,
                            float* __restrict__ out, int Mrows) {
    extern __shared__ float w3s[];   // [128 k-pairs][256 n][2] = 65536 floats = 256KB

    // ---- stage W3 -> LDS (all 256 threads participate, before any guard)
    for (int p = threadIdx.x; p < 128 * 256; p += 256) {
        int kk = p >> 8;            // k-pair index
        int n  = p & 255;
        w3s[p * 2 + 0] = W3[(long)(2 * kk)     * HID + n];
        w3s[p * 2 + 1] = W3[(long)(2 * kk + 1) * HID + n];
    }
    __syncthreads();

    int wave   = (blockIdx.x * blockDim.x + threadIdx.x) >> 5;
    int chunks = Mrows >> 4;
    if (wave >= chunks) return;                                 // wave-uniform
    int m0   = wave << 4;
    int lane = threadIdx.x & 31;
    int half = lane >> 4;
    int r    = lane & 15;

    int rowA = m0 + r;                                          // A row this lane feeds
    long base0 = (long)li0[rowA] * OUTD;
    long base1 = (long)li1[rowA] * OUTD;

    v8f dacc[16];
#pragma unroll
    for (int nt = 0; nt < 16; ++nt) dacc[nt] = v8f{};

    for (int k = 0; k < 2 * OUTD; k += 4) {
        int ka = k + 2 * half;                                  // even; ka,ka+1 same side of 128
        long ab = (ka < OUTD) ? (base0 + ka) : (base1 + ka - OUTD);
        v2f a;
        a[0] = Z[ab];
        a[1] = Z[ab + 1];
        const v2f* brow = (const v2f*)(w3s + (long)(ka >> 1) * 512) + r;  // pair (ka,ka+1), col r
#pragma unroll
        for (int nt = 0; nt < 16; ++nt) {
            v2f b = brow[nt * 16];                              // ds_load_b64
            dacc[nt] = __builtin_amdgcn_wmma_f32_16x16x4_f32(false, a, false, b,
                                                             (short)0, dacc[nt], false, false);
        }
    }

    // ---- epilogue: relu + b3, dot with W4, reduce across the 16 lanes of each half
    float accs[8];
#pragma unroll
    for (int i = 0; i < 8; ++i) accs[i] = 0.0f;
#pragma unroll
    for (int nt = 0; nt < 16; ++nt) {
        int n0 = nt << 4;
        float w4 = W4[n0 + r];
        float bb = b3[n0 + r];
#pragma unroll
        for (int i = 0; i < 8; ++i)
            accs[i] += fmaxf(dacc[nt][i] + bb, 0.0f) * w4;      // row m0+i+8*half, col n0+r
    }
    float b4v = b4[0];
#pragma unroll
    for (int i = 0; i < 8; ++i) {
        float v = accs[i];
        for (int m = 1; m < 16; m <<= 1) v += __shfl_xor(v, m, 32);
        if (r == 0) out[m0 + i + 8 * half] = v + b4v;
    }
}

// ---------------------------------------------------------------- launch
extern "C" void kernel_launch(void* const* d_in, const int* in_sizes, int n_in,
                              void* d_out, int out_size, void* d_ws, size_t ws_size,
                              hipStream_t stream) {
    const float* x   = (const float*)d_in[0];
    const int*   ei  = (const int*)d_in[1];   // [2, NE]
    const int*   eli = (const int*)d_in[2];   // [2, NL]
    const float* W1  = (const float*)d_in[3];
    const float* b1  = (const float*)d_in[4];
    const float* W2  = (const float*)d_in[5];
    const float* b2  = (const float*)d_in[6];
    const float* W3  = (const float*)d_in[7];
    const float* b3  = (const float*)d_in[8];
    const float* W4  = (const float*)d_in[9];
    const float* b4  = (const float*)d_in[10];
    float* out = (float*)d_out;

    const int* src = ei;
    const int* dst = ei + NE;
    const int* li0 = eli;
    const int* li1 = eli + NL;

    // workspace layout (floats): deg | dinv | invdeg | B0 (NN*256) | B1 (NN*256)
    float* deg    = (float*)d_ws;
    float* dinv   = deg + NN;
    float* invdeg = dinv + NN;
    float* B0     = invdeg + NN;
    float* B1     = B0 + (long)NN * HID;

    const int TB = 256;
    auto blk = [](long n, int t) { return (unsigned)((n + t - 1) / t); };

    // degrees
    zero_f32<<<blk(NN, TB), TB, 0, stream>>>(deg, NN);
    count_deg<<<blk(NE, TB), TB, 0, stream>>>(dst, deg, NE);
    finalize_deg<<<blk(NN, TB), TB, 0, stream>>>(deg, dinv, invdeg, NN);

    // conv1: H1 = x @ W1 -> B0 ; agg -> B1 ; Z1 = relu(agg + H1/deg + b1) in B1
    {
        long waves = ((long)NN / 16) * (HID / 64);
        gemm_wmma<<<blk(waves * 32, TB), TB, 0, stream>>>(x, W1, B0, NN, HID, IND);
        zero_f32<<<blk((long)NN * HID, TB), TB, 0, stream>>>(B1, (long)NN * HID);
        edge_scatter<<<blk((long)NE * (HID / 4), TB), TB, 0, stream>>>(B0, src, dst, dinv, B1, NE, HID);
        post_conv<<<blk((long)NN * HID, TB), TB, 0, stream>>>(B1, B0, invdeg, b1, NN, HID, 1);
    }

    // conv2: H2 = Z1 @ W2 -> B0 ; agg -> B1 (Z1 dead) ; Z2 = agg + H2/deg + b2 in B1
    {
        long waves = ((long)NN / 16) * (OUTD / 64);
        gemm_wmma<<<blk(waves * 32, TB), TB, 0, stream>>>(B1, W2, B0, NN, OUTD, HID);
        zero_f32<<<blk((long)NN * OUTD, TB), TB, 0, stream>>>(B1, (long)NN * OUTD);
        edge_scatter<<<blk((long)NE * (OUTD / 4), TB), TB, 0, stream>>>(B0, src, dst, dinv, B1, NE, OUTD);
        post_conv<<<blk((long)NN * OUTD, TB), TB, 0, stream>>>(B1, B0, invdeg, b2, NN, OUTD, 0);
    }

    // fused decode: out = relu(concat(Z2[li0],Z2[li1]) @ W3 + b3) @ W4 + b4
    {
        long waves = NL / 16;                         // 62500 waves, 8 per block
        size_t ldsBytes = (size_t)128 * 256 * 2 * sizeof(float);  // 256KB dynamic LDS
        decode_wmma<<<blk(waves * 32, TB), TB, ldsBytes, stream>>>(B1, li0, li1, W3, b3, W4, b4,
                                                                   out, NL);
    }
}